// PolynomialToRoots_71511205479149
// MI455X (gfx1250) — compile-verified
//
#include <hip/hip_runtime.h>
#include <hip/hip_bf16.h>

// Polynomial root finding: one polynomial per wave32, one root per lane.
// Durand-Kerner simultaneous iteration:  z_i <- z_i - p(z_i)/prod_{j!=i}(z_i - z_j)
// Coefficients staged global->LDS with gfx1250 async copy (ASYNCcnt), then
// broadcast into registers; cross-lane gathers via independent ds_bpermute ops.

#define POLY_ORDER   32
#define NCOEF        33
#define WAVES_PER_BLOCK 8
#define BLOCK_THREADS   256
#define DK_ITERS     80
#define LDS_STRIDE   36   // pad 33 -> 36 floats per wave slot

__global__ __launch_bounds__(BLOCK_THREADS, 2)
void PolynomialToRoots_71511205479149_kernel(const float* __restrict__ a,
                                             float2* __restrict__ out,
                                             int batch)
{
    __shared__ float lds[WAVES_PER_BLOCK * LDS_STRIDE];

    const int lane = threadIdx.x & 31;
    const int wid  = threadIdx.x >> 5;
    const int poly = blockIdx.x * WAVES_PER_BLOCK + wid;
    if (poly >= batch) return;

    const float* ap   = a + (size_t)poly * NCOEF;
    float*       cbuf = &lds[wid * LDS_STRIDE];

    // ---- CDNA5 async global->LDS staging of the 33 coefficients ----------
    // LDS byte address == low 32 bits of the flat shared-memory pointer
    // (ISA 10.2 aperture mapping).
    {
        const float* gsrc = ap + lane;
        unsigned ldsDst = (unsigned)(unsigned long long)(const void*)(cbuf + lane);
        asm volatile("global_load_async_to_lds_b32 %0, %1, off"
                     :: "v"(ldsDst), "v"(gsrc) : "memory");
        if (lane == 0) {
            const float* gsrc2 = ap + 32;
            unsigned ldsDst2 = (unsigned)(unsigned long long)(const void*)(cbuf + 32);
            asm volatile("global_load_async_to_lds_b32 %0, %1, off"
                         :: "v"(ldsDst2), "v"(gsrc2) : "memory");
        }
        asm volatile("s_wait_asynccnt 0" ::: "memory");
    }

    // ---- Normalize to monic form; replicate all coefficients per lane ----
    // p(z) = z^32 + b1*z^31 + ... + b32, b_k = a_k / a_0.
    const float inv_a0 = 1.0f / cbuf[0];
    float b[NCOEF];
    b[0] = 1.0f;
#pragma unroll
    for (int k = 1; k < NCOEF; ++k) b[k] = cbuf[k] * inv_a0;

    // ---- Initial guesses: circle at geometric-mean root radius -----------
    // |prod roots| = |b32|  =>  R ~ |b32|^(1/32); angle offset breaks
    // real-axis symmetry so no start sits on a root-symmetry fixed point.
    const float TWO_PI = 6.28318530717958647692f;
    float R = __powf(fmaxf(fabsf(b[32]), 1e-20f), 1.0f / 32.0f);
    R = fminf(fmaxf(R, 0.5f), 2.0f);
    const float th = TWO_PI * ((float)lane + 0.3530f) * (1.0f / 32.0f);
    float zr = R * __cosf(th);
    float zi = R * __sinf(th);

    // ---- Fixed-count Durand-Kerner iteration (deterministic) -------------
#pragma unroll 1
    for (int it = 0; it < DK_ITERS; ++it) {
        // Numerator p(z): two half-Horner chains combined as H1*z^16 + H2
        // (halves the serial FMA depth vs a single 32-step Horner).
        float h1r = 1.0f, h1i = 0.0f;          // b0..b16  -> degree-16 chain
        float h2r = b[17], h2i = 0.0f;         // b17..b32 -> degree-15 chain
#pragma unroll
        for (int k = 1; k <= 16; ++k) {
            float t1r = fmaf(h1r, zr, fmaf(-h1i, zi, b[k]));
            float t1i = fmaf(h1r, zi,  h1i * zr);
            h1r = t1r; h1i = t1i;
            if (k <= 15) {
                float t2r = fmaf(h2r, zr, fmaf(-h2i, zi, b[17 + k]));
                float t2i = fmaf(h2r, zi,  h2i * zr);
                h2r = t2r; h2i = t2i;
            }
        }
        // z^16 via 4 complex squarings (independent of both Horner chains)
        float sr = zr, si = zi;
#pragma unroll
        for (int q = 0; q < 4; ++q) {
            float tr = sr * sr - si * si;
            float ti = 2.0f * sr * si;
            sr = tr; si = ti;
        }
        const float pr = fmaf(h1r, sr, fmaf(-h1i, si, h2r));
        const float pi = fmaf(h1r, si, fmaf( h1i, sr, h2i));

        // Denominator prod_{j != i} (z_i - z_j):
        // 31 INDEPENDENT ds_bpermute gathers from the original z (index
        // (lane+s)&31), accumulated into two alternating complex-product
        // chains to halve multiply depth; combined at the end.
        float d0r = 1.0f, d0i = 0.0f;
        float d1r = 1.0f, d1i = 0.0f;
#pragma unroll
        for (int s = 1; s < 32; ++s) {
            const int idx = (lane + s) & 31;
            const float wr = __shfl(zr, idx, 32);
            const float wi = __shfl(zi, idx, 32);
            const float er = zr - wr;
            const float ei = zi - wi;
            if (s & 1) {
                float tr = d0r * er - d0i * ei;
                float ti = fmaf(d0r, ei, d0i * er);
                d0r = tr; d0i = ti;
            } else {
                float tr = d1r * er - d1i * ei;
                float ti = fmaf(d1r, ei, d1i * er);
                d1r = tr; d1i = ti;
            }
        }
        const float dr = d0r * d1r - d0i * d1i;
        const float di = fmaf(d0r, d1i, d0i * d1r);

        // z -= p/d   (complex divide via conjugate; eps keeps it finite)
        const float inv = 1.0f / fmaf(dr, dr, fmaf(di, di, 1e-30f));
        zr -= (pr * dr + pi * di) * inv;
        zi -= (pi * dr - pr * di) * inv;
    }

    out[(size_t)poly * POLY_ORDER + lane] = make_float2(zr, zi);
}

extern "C" void kernel_launch(void* const* d_in, const int* in_sizes, int n_in,
                              void* d_out, int out_size, void* d_ws, size_t ws_size,
                              hipStream_t stream) {
    (void)n_in; (void)out_size; (void)d_ws; (void)ws_size;
    const float* a = (const float*)d_in[0];
    float2* out = (float2*)d_out;
    const int batch  = in_sizes[0] / NCOEF;                 // 32768
    const int blocks = (batch + WAVES_PER_BLOCK - 1) / WAVES_PER_BLOCK;
    PolynomialToRoots_71511205479149_kernel<<<blocks, BLOCK_THREADS, 0, stream>>>(a, out, batch);
}